// fftConv2d_54743653154849
// MI455X (gfx1250) — compile-verified
//
#include <hip/hip_runtime.h>

#define PI_F 3.14159265358979323846f

typedef float v2f __attribute__((ext_vector_type(2)));
typedef float v8f __attribute__((ext_vector_type(8)));

// ---------------------------------------------------------------------------
// 256-point Stockham radix-2 complex FFT in LDS.
// 128 threads, one butterfly per thread per stage, 8 stages, ping-pong
// between (rA,iA) and (rB,iB). Result lands back in (rA,iA), natural order.
// dir = -1: forward (exp(-i...)), dir = +1: inverse (exp(+i...), unscaled).
// ---------------------------------------------------------------------------
__device__ __forceinline__ void block_fft256(
    float* rA, float* iA, float* rB, float* iB, int tid, float dir)
{
  float *srcR = rA, *srcI = iA, *dstR = rB, *dstI = iB;
  int n = 256, s = 1;
#pragma unroll
  for (int stage = 0; stage < 8; ++stage) {
    int m = n >> 1;
    int p = tid / s;          // p in [0, m)   (s is a power of two literal per iter)
    int q = tid - p * s;      // q in [0, s)
    float theta = dir * (2.0f * PI_F) * (float)p / (float)n;
    float wr = __cosf(theta);
    float wi = __sinf(theta);
    float ar = srcR[q + s * p],       ai = srcI[q + s * p];
    float br = srcR[q + s * (p + m)], bi = srcI[q + s * (p + m)];
    dstR[q + s * (2 * p)] = ar + br;
    dstI[q + s * (2 * p)] = ai + bi;
    float tr = ar - br, ti = ai - bi;
    dstR[q + s * (2 * p + 1)] = tr * wr - ti * wi;
    dstI[q + s * (2 * p + 1)] = tr * wi + ti * wr;
    __syncthreads();
    float* t;
    t = srcR; srcR = dstR; dstR = t;
    t = srcI; srcI = dstI; dstI = t;
    n >>= 1; s <<= 1;
  }
}

// ---------------------------------------------------------------------------
// Stage 1: real row FFT over W. One row of 256 reals per block.
// row = bc*256 + h  (bc = b*C+c or o*C+c). Writes bins 0..128 transposed:
// out[(bc*129 + wb)*256 + h]  -> layout [bc][wb][h], h contiguous.
// ---------------------------------------------------------------------------
__global__ void __launch_bounds__(128)
fftconv_rows_r2c(const float* __restrict__ in, float2* __restrict__ out)
{
  __shared__ float rA[256], iA[256], rB[256], iB[256];
  const int tid = threadIdx.x;
  const int row = blockIdx.x;
  const float* src = in + (size_t)row * 256;
  rA[tid]       = src[tid];
  rA[tid + 128] = src[tid + 128];
  iA[tid] = 0.0f;
  iA[tid + 128] = 0.0f;
  __syncthreads();
  block_fft256(rA, iA, rB, iB, tid, -1.0f);
  const int bc = row >> 8;
  const int h  = row & 255;
  float2* o = out + (size_t)bc * 129 * 256 + h;
  for (int wb = tid; wb < 129; wb += 128)
    o[(size_t)wb * 256] = make_float2(rA[wb], iA[wb]);
}

// ---------------------------------------------------------------------------
// Stage 2/4: in-place complex FFT over H (contiguous 256 complex per column).
// col = bc*129 + wb. dir=-1 forward (scale=1), dir=+1 inverse (scale=1/256).
// ---------------------------------------------------------------------------
__global__ void __launch_bounds__(128)
fftconv_cols_c2c(float2* data, float dir, float scale)
{
  __shared__ float rA[256], iA[256], rB[256], iB[256];
  const int tid = threadIdx.x;
  float2* p = data + (size_t)blockIdx.x * 256;
  float2 v0 = p[tid];
  float2 v1 = p[tid + 128];
  rA[tid] = v0.x;       iA[tid] = v0.y;
  rA[tid + 128] = v1.x; iA[tid + 128] = v1.y;
  __syncthreads();
  block_fft256(rA, iA, rB, iB, tid, dir);
  p[tid]       = make_float2(rA[tid] * scale,       iA[tid] * scale);
  p[tid + 128] = make_float2(rA[tid + 128] * scale, iA[tid + 128] * scale);
}

// ---------------------------------------------------------------------------
// Stage 3: per-frequency-bin complex GEMM  Y[b][o] = sum_c X[b][c] * K[c][o]
// (16x32 @ 32x32 complex) using V_WMMA_F32_16X16X4_F32. One wave per bin.
// Xf/Yf layout: [(b*32+c)][wb][h]; Kf layout: [(o*32+c)][wb][h].
// In-place: Yf == Xf is safe (all X loads precede stores within the wave,
// and each bin owns a disjoint (wb,h) slice of the buffer).
// ---------------------------------------------------------------------------
__global__ void __launch_bounds__(256)
fftconv_bins_wmma(const float2* Xf, const float2* __restrict__ Kf, float2* Yf)
{
  const int C = 32;
  const size_t CH = (size_t)129 * 256;      // elements per (b,c) plane

  const int lane = threadIdx.x & 31;
  const int wave = threadIdx.x >> 5;
  const int bin  = blockIdx.x * 8 + wave;   // nbins = 129*256 = 33024 (mult of 8)
  if (bin >= 129 * 256) return;
  const int wb = bin >> 8;
  const int h  = bin & 255;
  const size_t binoff = (size_t)wb * 256 + h;

  const int bM    = lane & 15;   // A-matrix row (batch b)
  const int khalf = lane >> 4;   // K-slot half: lanes 0-15 -> K 0..1, 16-31 -> K 2..3
  const int oN    = lane & 15;   // B/D-matrix column (output channel o)

  // Preload the full A (X) tile for this bin: 8 K-steps x 2 regs, complex.
  float xr[8][2], xi[8][2];
#pragma unroll
  for (int kk = 0; kk < 8; ++kk) {
#pragma unroll
    for (int v = 0; v < 2; ++v) {
      const int c = 4 * kk + 2 * khalf + v;
      float2 t = Xf[(size_t)(bM * C + c) * CH + binoff];
      xr[kk][v] = t.x;
      xi[kk][v] = t.y;
    }
  }

  v8f accR0 = {}, accI0 = {}, accR1 = {}, accI1 = {};
#pragma unroll
  for (int kk = 0; kk < 8; ++kk) {
    v2f aR, aI, aIn;
    aR[0] = xr[kk][0];  aR[1] = xr[kk][1];
    aI[0] = xi[kk][0];  aI[1] = xi[kk][1];
    aIn[0] = -xi[kk][0]; aIn[1] = -xi[kk][1];

    v2f bR0, bI0, bR1, bI1;
#pragma unroll
    for (int v = 0; v < 2; ++v) {
      const int c = 4 * kk + 2 * khalf + v;
      float2 t0 = Kf[(size_t)(oN * C + c) * CH + binoff];          // o in 0..15
      float2 t1 = Kf[(size_t)((oN + 16) * C + c) * CH + binoff];   // o in 16..31
      bR0[v] = t0.x; bI0[v] = t0.y;
      bR1[v] = t1.x; bI1[v] = t1.y;
    }

    // Yr = Xr*Kr + (-Xi)*Ki ; Yi = Xr*Ki + Xi*Kr
    accR0 = __builtin_amdgcn_wmma_f32_16x16x4_f32(false, aR,  false, bR0, (short)0, accR0, false, false);
    accR0 = __builtin_amdgcn_wmma_f32_16x16x4_f32(false, aIn, false, bI0, (short)0, accR0, false, false);
    accI0 = __builtin_amdgcn_wmma_f32_16x16x4_f32(false, aR,  false, bI0, (short)0, accI0, false, false);
    accI0 = __builtin_amdgcn_wmma_f32_16x16x4_f32(false, aI,  false, bR0, (short)0, accI0, false, false);

    accR1 = __builtin_amdgcn_wmma_f32_16x16x4_f32(false, aR,  false, bR1, (short)0, accR1, false, false);
    accR1 = __builtin_amdgcn_wmma_f32_16x16x4_f32(false, aIn, false, bI1, (short)0, accR1, false, false);
    accI1 = __builtin_amdgcn_wmma_f32_16x16x4_f32(false, aR,  false, bI1, (short)0, accI1, false, false);
    accI1 = __builtin_amdgcn_wmma_f32_16x16x4_f32(false, aI,  false, bR1, (short)0, accI1, false, false);
  }

  // D layout: VGPR v holds row M=v (lanes 0-15) / M=v+8 (lanes 16-31), N=lane&15.
#pragma unroll
  for (int v = 0; v < 8; ++v) {
    const int b = v + 8 * khalf;
    Yf[(size_t)(b * C + oN) * CH + binoff]        = make_float2(accR0[v], accI0[v]);
    Yf[(size_t)(b * C + (oN + 16)) * CH + binoff] = make_float2(accR1[v], accI1[v]);
  }
}

// ---------------------------------------------------------------------------
// Stage 5: inverse row FFT over W with Hermitian reconstruction -> real out.
// row = bo*256 + h; gathers 129 bins (stride 256 complex), mirrors to 256,
// inverse FFT, writes out[bo][h][0..255] * (1/256), coalesced.
// ---------------------------------------------------------------------------
__global__ void __launch_bounds__(128)
fftconv_rows_c2r(const float2* __restrict__ Yf, float* __restrict__ out)
{
  __shared__ float rA[256], iA[256], rB[256], iB[256];
  const int tid = threadIdx.x;
  const int row = blockIdx.x;
  const int bo = row >> 8;
  const int h  = row & 255;
  const float2* src = Yf + (size_t)bo * 129 * 256 + h;
  for (int wb = tid; wb < 129; wb += 128) {
    float2 t = src[(size_t)wb * 256];
    rA[wb] = t.x;
    iA[wb] = t.y;
    if (wb >= 1 && wb <= 127) {        // Hermitian mirror; DC & Nyquist fixed
      rA[256 - wb] = t.x;
      iA[256 - wb] = -t.y;
    }
  }
  __syncthreads();
  block_fft256(rA, iA, rB, iB, tid, +1.0f);
  const float s = 1.0f / 256.0f;
  float* o = out + (size_t)row * 256;
  o[tid]       = rA[tid] * s;
  o[tid + 128] = rA[tid + 128] * s;
}

// ---------------------------------------------------------------------------
extern "C" void kernel_launch(void* const* d_in, const int* in_sizes, int n_in,
                              void* d_out, int out_size, void* d_ws, size_t ws_size,
                              hipStream_t stream)
{
  (void)in_sizes; (void)n_in; (void)out_size; (void)ws_size;
  const float* x = (const float*)d_in[0];   // (16, 32, 256, 256) f32
  const float* w = (const float*)d_in[1];   // (32, 32, 256, 256) f32
  float* out = (float*)d_out;               // (16, 32, 256, 256) f32

  const int B = 16, C = 32, O = 32, H = 256, WB = 129;

  // Workspace: Xf (also reused as Yf) then Kf. Needs ~406 MB.
  float2* Xf = (float2*)d_ws;
  float2* Kf = Xf + (size_t)B * C * WB * H;

  // Forward: rows (r2c, transposed out) then columns (c2c, in place).
  fftconv_rows_r2c<<<B * C * H, 128, 0, stream>>>(x, Xf);
  fftconv_rows_r2c<<<O * C * H, 128, 0, stream>>>(w, Kf);
  fftconv_cols_c2c<<<B * C * WB, 128, 0, stream>>>(Xf, -1.0f, 1.0f);
  fftconv_cols_c2c<<<O * C * WB, 128, 0, stream>>>(Kf, -1.0f, 1.0f);

  // Per-bin complex channel contraction on the matrix pipe (in place).
  const int nbins = WB * H;                 // 33024, multiple of 8 waves/block
  fftconv_bins_wmma<<<nbins / 8, 256, 0, stream>>>(Xf, Kf, Xf);

  // Inverse: columns (c2c, conj twiddles, 1/256) then rows (c2r, 1/256).
  fftconv_cols_c2c<<<B * O * WB, 128, 0, stream>>>(Xf, +1.0f, 1.0f / 256.0f);
  fftconv_rows_c2r<<<B * O * H, 128, 0, stream>>>(Xf, out);
}